// PrePromptMoE_21655225106530
// MI455X (gfx1250) — compile-verified
//
#include <hip/hip_runtime.h>
#include <math.h>

typedef __attribute__((ext_vector_type(2))) float v2f;
typedef __attribute__((ext_vector_type(4))) float v4f;
typedef __attribute__((ext_vector_type(8))) float v8f;

#define DDIM 768
#define NEXP 8
#define NQ   32
#define ROW  (NQ * DDIM)   // 24576 floats per token

// ---------------------------------------------------------------------------
// Kernel 1: gating logits via V_WMMA_F32_16X16X4_F32 (exact f32 GEMM).
// One wave (32 lanes) computes a 16-token x 16-expert tile (experts 8..15
// zero-padded). K=768 chained through the f32 accumulator in steps of 4.
//
// A (16x4 f32): lane L holds token M=L&15; VGPR0/1 hold K = 2*(L>>4), +1.
// B (4x16 f32): lane L holds col  N=L&15; VGPR0/1 hold K = 2*(L>>4), +1.
// D (16x16 f32): VGPR r, lane L -> (M = r + 8*(L>>4), N = L&15).
// ---------------------------------------------------------------------------
__global__ void moe_gate_logits_wmma(const float* __restrict__ x,
                                     const float* __restrict__ gate_w,
                                     float* __restrict__ logits, int T) {
    int wave = threadIdx.x >> 5;
    int lane = threadIdx.x & 31;
    int tile = blockIdx.x * (blockDim.x >> 5) + wave;
    int t0 = tile * 16;
    if (t0 >= T) return;

    int half = lane >> 4;     // 0 or 1 -> K subgroup
    int lm   = lane & 15;     // token row for A, expert col for B/D

    const float* xrow = x + (size_t)(t0 + lm) * DDIM + 2 * half;
    // all lanes load valid memory; lanes with lm>=NEXP zero their B operand
    float msk = (lm < NEXP) ? 1.0f : 0.0f;
    const float* grow = gate_w + (size_t)(lm & (NEXP - 1)) * DDIM + 2 * half;

    v8f acc = {};
    for (int k = 0; k < DDIM; k += 4) {
        v2f a, b;
        a.x = xrow[k];
        a.y = xrow[k + 1];
        b.x = grow[k] * msk;
        b.y = grow[k + 1] * msk;
        acc = __builtin_amdgcn_wmma_f32_16x16x4_f32(
            /*neg_a=*/false, a, /*neg_b=*/false, b,
            /*c_mod=*/(short)0, acc, /*reuse_a=*/false, /*reuse_b=*/false);
    }

    if (lm < NEXP) {
        for (int r = 0; r < 8; ++r) {
            int t = t0 + r + 8 * half;
            logits[(size_t)t * NEXP + lm] = acc[r];
        }
    }
}

// ---------------------------------------------------------------------------
// Kernel 2: per-token softmax over E=8, top-2, normalized weights, and
// LDS-reduced global accumulation of imp[E] (prob sums) and cnt[E].
// ---------------------------------------------------------------------------
__global__ void moe_softmax_top2(const float* __restrict__ logits,
                                 float* __restrict__ wgate,   // [T] x (w0,w1,g0,g1)
                                 int* __restrict__ gate_out,  // [T,2] in d_out
                                 float* __restrict__ impAcc,  // [E]
                                 int* __restrict__ cntAcc,    // [E]
                                 int T) {
    __shared__ float simp[NEXP];
    __shared__ int   scnt[NEXP];
    if (threadIdx.x < NEXP) { simp[threadIdx.x] = 0.0f; scnt[threadIdx.x] = 0; }
    __syncthreads();

    int t = blockIdx.x * blockDim.x + threadIdx.x;
    if (t < T) {
        float l[NEXP], p[NEXP];
        float m = -3.402823466e+38f;
        for (int e = 0; e < NEXP; ++e) {
            l[e] = logits[(size_t)t * NEXP + e];
            m = fmaxf(m, l[e]);
        }
        float s = 0.0f;
        for (int e = 0; e < NEXP; ++e) { p[e] = expf(l[e] - m); s += p[e]; }
        float inv = 1.0f / s;
        for (int e = 0; e < NEXP; ++e) p[e] *= inv;

        // top-2, lowest index wins ties (matches lax.top_k)
        int g0 = 0; float v0 = p[0];
        for (int e = 1; e < NEXP; ++e) if (p[e] > v0) { v0 = p[e]; g0 = e; }
        int g1 = (g0 == 0) ? 1 : 0; float v1 = p[g1];
        for (int e = 0; e < NEXP; ++e)
            if (e != g0 && p[e] > v1) { v1 = p[e]; g1 = e; }

        float wsum = v0 + v1;
        float w0 = v0 / wsum, w1 = v1 / wsum;

        gate_out[t * 2]     = g0;
        gate_out[t * 2 + 1] = g1;

        v4f wg;
        wg.x = w0; wg.y = w1;
        wg.z = __int_as_float(g0); wg.w = __int_as_float(g1);
        ((v4f*)wgate)[t] = wg;

        for (int e = 0; e < NEXP; ++e) atomicAdd(&simp[e], p[e]);
        atomicAdd(&scnt[g0], 1);
        atomicAdd(&scnt[g1], 1);
    }
    __syncthreads();
    if (threadIdx.x < NEXP) {
        atomicAdd(&impAcc[threadIdx.x], simp[threadIdx.x]);
        atomicAdd(&cntAcc[threadIdx.x], scnt[threadIdx.x]);
    }
}

// ---------------------------------------------------------------------------
// Kernel 3: scalar losses + gate_load (E=8 -> one thread suffices).
// ---------------------------------------------------------------------------
__global__ void moe_finalize(const float* __restrict__ impAcc,
                             const int* __restrict__ cntAcc,
                             float* __restrict__ balOut,
                             float* __restrict__ impOut,
                             int* __restrict__ gate_load, int T) {
    if (threadIdx.x == 0 && blockIdx.x == 0) {
        float imp[NEXP]; int cnt[NEXP];
        float impSum = 0.0f; long tot = 0;
        for (int e = 0; e < NEXP; ++e) {
            imp[e] = impAcc[e]; cnt[e] = cntAcc[e];
            impSum += imp[e]; tot += cnt[e];
        }
        float fT = (float)T, fTot = (float)tot;
        float bal = 0.0f;
        for (int e = 0; e < NEXP; ++e)
            bal += (imp[e] / fT) * ((float)cnt[e] / fTot);
        bal *= (float)NEXP;

        float mu = impSum / (float)NEXP;
        float var = 0.0f;
        for (int e = 0; e < NEXP; ++e) { float d = imp[e] - mu; var += d * d; }
        var /= (float)(NEXP - 1);                 // ddof=1

        *balOut = bal;
        *impOut = var / (mu * mu);
        for (int e = 0; e < NEXP; ++e) gate_load[e] = cnt[e];
    }
}

// ---------------------------------------------------------------------------
// Kernel 4: the bandwidth phase. One block per token:
// out[t] = w0*cand[g0] + w1*cand[g1], float4 streams, NT stores so the
// 402 MB output stream doesn't evict the L2-resident cand (786 KB).
// ---------------------------------------------------------------------------
__global__ void moe_broadcast(const float* __restrict__ wgate,
                              const float* __restrict__ cand,
                              float* __restrict__ out) {
    int t = blockIdx.x;
    v4f wg = ((const v4f*)wgate)[t];
    float w0 = wg.x, w1 = wg.y;
    int g0 = __float_as_int(wg.z);
    int g1 = __float_as_int(wg.w);

    const v4f* c0 = (const v4f*)(cand + (size_t)g0 * ROW);
    const v4f* c1 = (const v4f*)(cand + (size_t)g1 * ROW);
    v4f* orow = (v4f*)(out + (size_t)t * ROW);

    for (int i = threadIdx.x; i < ROW / 4; i += blockDim.x) {
        v4f a = c0[i];
        v4f b = c1[i];
        v4f r = w0 * a + w1 * b;
        __builtin_nontemporal_store(r, &orow[i]);
    }
}

// ---------------------------------------------------------------------------
extern "C" void kernel_launch(void* const* d_in, const int* in_sizes, int n_in,
                              void* d_out, int out_size, void* d_ws, size_t ws_size,
                              hipStream_t stream) {
    const float* x      = (const float*)d_in[0];   // [B,S,D]
    const float* gate_w = (const float*)d_in[1];   // [E,D]
    const float* cand   = (const float*)d_in[2];   // [E,Nq,D]
    int T = in_sizes[0] / DDIM;                    // 4096

    // d_out tuple layout: out [T*NQ*D] f32 | bal | impLoss | gate_load[E] i32 | gate[T*2] i32
    float* out       = (float*)d_out;
    size_t outElems  = (size_t)T * ROW;
    float* balOut    = out + outElems;
    float* impOut    = balOut + 1;
    int*   gate_load = (int*)(impOut + 1);
    int*   gate      = gate_load + NEXP;

    // workspace layout
    float* logits = (float*)d_ws;                      // T*E floats
    float* wgate  = logits + (size_t)T * NEXP;         // T*4 floats (16B aligned)
    float* impAcc = wgate + (size_t)T * 4;             // E floats
    int*   cntAcc = (int*)(impAcc + NEXP);             // E ints

    // deterministic zero of atomic accumulators (graph-capturable)
    hipMemsetAsync(impAcc, 0, NEXP * sizeof(float) + NEXP * sizeof(int), stream);

    // 1) gating GEMM via WMMA: 8 waves/block, 16 tokens per wave
    int tiles = T / 16;                                // 256
    int wavesPerBlock = 8;
    moe_gate_logits_wmma<<<tiles / wavesPerBlock, 32 * wavesPerBlock, 0, stream>>>(
        x, gate_w, logits, T);

    // 2) softmax + top-2 + stats
    moe_softmax_top2<<<(T + 255) / 256, 256, 0, stream>>>(
        logits, wgate, gate, impAcc, cntAcc, T);

    // 3) losses + gate_load
    moe_finalize<<<1, 32, 0, stream>>>(impAcc, cntAcc, balOut, impOut, gate_load, T);

    // 4) bandwidth-bound broadcast (one block per token)
    moe_broadcast<<<T, 256, 0, stream>>>(wgate, cand, out);
}